// MolEncoderGAT_50603304681670
// MI455X (gfx1250) — compile-verified
//
#include <hip/hip_runtime.h>

// MolEncoderGAT for MI455X (gfx1250, wave32).
// Structure:
//   layer1 aggregation factored through the 9-dim input (36 atomics/edge),
//   fused dense-expand + bias + ELU + layer2 attention dots,
//   WMMA f32 16x16x4 GEMM for h1act[N,512] @ W2[512,128] with
//   GLOBAL_LOAD_ASYNC_TO_LDS_B128 staging of W2 chunks (ASYNCcnt path),
//   atomic segment-softmax (max / sum / scatter) per layer, atomic mean pool.

#define IN_F   9
#define HID    128
#define HEADS  4
#define L1OUT  (HEADS * HID)   // 512
#define NEG_SLOPE 0.2f

typedef __attribute__((ext_vector_type(2))) float v2f;
typedef __attribute__((ext_vector_type(8))) float v8f;

__device__ __forceinline__ float lrelu(float x) { return x > 0.f ? x : NEG_SLOPE * x; }
__device__ __forceinline__ float eluf(float x)  { return x > 0.f ? x : (expf(x) - 1.f); }

// float atomic max via monotone int mapping (memory initialized to -1e30f)
__device__ __forceinline__ void atomicMaxF(float* addr, float v) {
    if (v >= 0.f) atomicMax((int*)addr, __float_as_int(v));
    else          atomicMin((unsigned int*)addr, __float_as_uint(v));
}

// ---------------------------------------------------------------- fills
__global__ void k_fill(float* p, long long n, float v) {
    long long i = (long long)blockIdx.x * blockDim.x + threadIdx.x;
    long long stride = (long long)gridDim.x * blockDim.x;
    for (; i < n; i += stride) p[i] = v;
}

// ------------------------------------------------- precompute attention vectors
// ws1[h][i] = sum_c W1[i, h*128+c] * a_src1[h,c]   (and wd1 with a_dst1)
// w2s[j]    = sum_c W2[j, c] * a_src2[c]           (and w2d with a_dst2)
__global__ void k_prep(const float* __restrict__ W1, const float* __restrict__ as1,
                       const float* __restrict__ ad1, const float* __restrict__ W2,
                       const float* __restrict__ as2, const float* __restrict__ ad2,
                       float* __restrict__ ws1, float* __restrict__ wd1,
                       float* __restrict__ w2s, float* __restrict__ w2d) {
    int t = threadIdx.x;
    if (t < HEADS * IN_F) {
        int h = t / IN_F, i = t % IN_F;
        float s = 0.f, d = 0.f;
        for (int c = 0; c < HID; ++c) {
            float w = W1[i * L1OUT + h * HID + c];
            s += w * as1[h * HID + c];
            d += w * ad1[h * HID + c];
        }
        ws1[t] = s; wd1[t] = d;
    }
    if (t < L1OUT) {
        float s = 0.f, d = 0.f;
        for (int c = 0; c < HID; ++c) {
            float w = W2[t * HID + c];
            s += w * as2[c];
            d += w * ad2[c];
        }
        w2s[t] = s; w2d[t] = d;
    }
}

// ------------------------------------------------- per-node layer1 logits
__global__ void k_alpha1(const float* __restrict__ x, const float* __restrict__ ws1,
                         const float* __restrict__ wd1, float* __restrict__ as1n,
                         float* __restrict__ ad1n, int N) {
    int n = blockIdx.x * blockDim.x + threadIdx.x;
    if (n >= N) return;
    float xv[IN_F];
#pragma unroll
    for (int i = 0; i < IN_F; ++i) xv[i] = x[n * IN_F + i];
#pragma unroll
    for (int h = 0; h < HEADS; ++h) {
        float s = 0.f, d = 0.f;
#pragma unroll
        for (int i = 0; i < IN_F; ++i) {
            s += xv[i] * ws1[h * IN_F + i];
            d += xv[i] * wd1[h * IN_F + i];
        }
        as1n[n * HEADS + h] = s;
        ad1n[n * HEADS + h] = d;
    }
}

// ------------------------------------------------- edge passes (max, sum)
__global__ void k_edge_max(const int* __restrict__ ei, int E, int N,
                           const float* __restrict__ as, const float* __restrict__ ad,
                           float* __restrict__ m, int H) {
    int e = blockIdx.x * blockDim.x + threadIdx.x;
    if (e >= E + N) return;
    int s = (e < E) ? ei[e]     : (e - E);
    int d = (e < E) ? ei[E + e] : (e - E);
    for (int h = 0; h < H; ++h)
        atomicMaxF(&m[d * H + h], lrelu(as[s * H + h] + ad[d * H + h]));
}

__global__ void k_edge_sum(const int* __restrict__ ei, int E, int N,
                           const float* __restrict__ as, const float* __restrict__ ad,
                           const float* __restrict__ m, float* __restrict__ z, int H) {
    int e = blockIdx.x * blockDim.x + threadIdx.x;
    if (e >= E + N) return;
    int s = (e < E) ? ei[e]     : (e - E);
    int d = (e < E) ? ei[E + e] : (e - E);
    for (int h = 0; h < H; ++h) {
        float l = lrelu(as[s * H + h] + ad[d * H + h]);
        atomicAdd(&z[d * H + h], expf(l - m[d * H + h]));
    }
}

// ------------------------------------------------- layer1 factored scatter
// xagg[dst, h, i] += alpha_eh * x[src, i]   (36 atomics / edge)
__global__ void k_scatter1(const int* __restrict__ ei, int E, int N,
                           const float* __restrict__ x, const float* __restrict__ as,
                           const float* __restrict__ ad, const float* __restrict__ m,
                           const float* __restrict__ z, float* __restrict__ xagg) {
    int e = blockIdx.x * blockDim.x + threadIdx.x;
    if (e >= E + N) return;
    int s = (e < E) ? ei[e]     : (e - E);
    int d = (e < E) ? ei[E + e] : (e - E);
    float w[HEADS];
#pragma unroll
    for (int h = 0; h < HEADS; ++h) {
        float l = lrelu(as[s * HEADS + h] + ad[d * HEADS + h]);
        w[h] = expf(l - m[d * HEADS + h]) / (z[d * HEADS + h] + 1e-16f);
    }
    float xv[IN_F];
#pragma unroll
    for (int i = 0; i < IN_F; ++i) xv[i] = x[s * IN_F + i];
#pragma unroll
    for (int h = 0; h < HEADS; ++h)
#pragma unroll
        for (int i = 0; i < IN_F; ++i)
            atomicAdd(&xagg[((long long)d * HEADS + h) * IN_F + i], w[h] * xv[i]);
}

// ------------------------------------------------- layer1 dense expand + ELU + layer2 dots
// h1act[n, j] = elu( xagg[n, j>>7, :] . W1[:, j] + b1[j] )
// as2[n] = h1act[n,:] . w2s ; ad2[n] = h1act[n,:] . w2d
__global__ void k_dense1(const float* __restrict__ xagg, const float* __restrict__ W1,
                         const float* __restrict__ b1, const float* __restrict__ w2s,
                         const float* __restrict__ w2d, float* __restrict__ h1act,
                         float* __restrict__ as2n, float* __restrict__ ad2n, int N) {
    __shared__ float W1s[IN_F * L1OUT];   // 18 KB
    for (int i = threadIdx.x; i < IN_F * L1OUT; i += blockDim.x) W1s[i] = W1[i];
    __syncthreads();
    int wave = threadIdx.x >> 5, lane = threadIdx.x & 31;
    int n = blockIdx.x * (blockDim.x >> 5) + wave;
    if (n >= N) return;
    float xa[HEADS][IN_F];
#pragma unroll
    for (int h = 0; h < HEADS; ++h)
#pragma unroll
        for (int i = 0; i < IN_F; ++i)
            xa[h][i] = xagg[((long long)n * HEADS + h) * IN_F + i];
    float ss = 0.f, dd = 0.f;
#pragma unroll
    for (int t = 0; t < 16; ++t) {
        int j = lane + 32 * t;
        int h = j >> 7;
        float v = 0.f;
#pragma unroll
        for (int i = 0; i < IN_F; ++i) v += xa[h][i] * W1s[i * L1OUT + j];
        v = eluf(v + b1[j]);
        h1act[(long long)n * L1OUT + j] = v;
        ss += v * w2s[j];
        dd += v * w2d[j];
    }
#pragma unroll
    for (int off = 16; off >= 1; off >>= 1) {
        ss += __shfl_xor(ss, off, 32);
        dd += __shfl_xor(dd, off, 32);
    }
    if (lane == 0) { as2n[n] = ss; ad2n[n] = dd; }
}

// ------------------------------------------------- layer2 GEMM via WMMA f32 16x16x4
// C[N,128] = A[N,512] @ W2[512,128]; one wave owns a 16x128 strip.
// W2 K-chunks (64 KB, contiguous in global) are staged into LDS with
// GLOBAL_LOAD_ASYNC_TO_LDS_B128 (ASYNCcnt-tracked async copy, no VGPR staging).
__global__ void __launch_bounds__(256) k_gemm2(const float* __restrict__ A,
                                               const float* __restrict__ W2,
                                               float* __restrict__ C, int N) {
    __shared__ float Bs[128 * HID];       // 64 KB K-chunk of W2
    int wave = threadIdx.x >> 5, lane = threadIdx.x & 31;
    int row0 = blockIdx.x * 128 + wave * 16;
    int mlo  = lane & 15;
    int khalf = (lane >> 4) << 1;         // lanes 0-15 -> K+0/K+1, lanes 16-31 -> K+2/K+3
    v8f acc[8] = {};

    unsigned long long w2base = (unsigned long long)(const void*)W2;

    for (int kb = 0; kb < L1OUT; kb += 128) {
        // async copy: chunk is rows kb..kb+127 (all 128 cols) = 64 KB contiguous.
        // 4096 B128 transfers; each lane moves 16 B global -> 16 B LDS.
        for (int q = threadIdx.x; q < (128 * HID) / 4; q += 256) {
            unsigned int voff = (unsigned int)(kb * HID * 4 + q * 16); // global byte offset
            // low 32 bits of the flat pointer == wave-relative LDS byte address
            unsigned int ldsa = (unsigned int)(unsigned long long)(const void*)&Bs[q * 4];
            asm volatile("global_load_async_to_lds_b128 %0, %1, %2"
                         :
                         : "v"(ldsa), "v"(voff), "s"(w2base)
                         : "memory");
        }
        asm volatile("s_wait_asynccnt 0" ::: "memory");
        __syncthreads();

        int r = row0 + mlo; if (r >= N) r = N - 1;   // branchless clamp: EXEC stays all-ones
        const float* Arow = A + (long long)r * L1OUT + kb;
        for (int kk = 0; kk < 128; kk += 4) {
            v2f a;
            a.x = Arow[kk + khalf];
            a.y = Arow[kk + khalf + 1];
#pragma unroll
            for (int t = 0; t < 8; ++t) {
                v2f b;
                b.x = Bs[(kk + khalf) * HID + t * 16 + mlo];
                b.y = Bs[(kk + khalf + 1) * HID + t * 16 + mlo];
                acc[t] = __builtin_amdgcn_wmma_f32_16x16x4_f32(
                    false, a, false, b, (short)0, acc[t], false, false);
            }
        }
        __syncthreads();
    }
    int mbase = (lane >> 4) << 3;         // C layout: vgpr r -> M=r (lanes 0-15) / M=r+8
#pragma unroll
    for (int t = 0; t < 8; ++t)
#pragma unroll
        for (int r8 = 0; r8 < 8; ++r8) {
            int grow = row0 + mbase + r8;
            if (grow < N) C[(long long)grow * HID + t * 16 + mlo] = acc[t][r8];
        }
}

// ------------------------------------------------- layer2 scatter (wave per edge)
__global__ void k_scatter2(const int* __restrict__ ei, int E, int N,
                           const float* __restrict__ h2lin, const float* __restrict__ as,
                           const float* __restrict__ ad, const float* __restrict__ m,
                           const float* __restrict__ z, float* __restrict__ agg2) {
    int wave = threadIdx.x >> 5, lane = threadIdx.x & 31;
    int e = blockIdx.x * (blockDim.x >> 5) + wave;
    if (e >= E + N) return;
    int s = (e < E) ? ei[e]     : (e - E);
    int d = (e < E) ? ei[E + e] : (e - E);
    float l = lrelu(as[s] + ad[d]);
    float w = expf(l - m[d]) / (z[d] + 1e-16f);
    const float* src = h2lin + (long long)s * HID;
    float* dst = agg2 + (long long)d * HID;
#pragma unroll
    for (int t = 0; t < 4; ++t) {
        int c = lane + 32 * t;
        atomicAdd(&dst[c], src[c] * w);
    }
}

// ------------------------------------------------- bias + ELU + mean-pool accumulate
__global__ void k_pool(const float* __restrict__ agg2, const float* __restrict__ b2,
                       const int* __restrict__ batch, float* __restrict__ gsum,
                       float* __restrict__ gcnt, int N) {
    int wave = threadIdx.x >> 5, lane = threadIdx.x & 31;
    int n = blockIdx.x * (blockDim.x >> 5) + wave;
    if (n >= N) return;
    int g = batch[n];
#pragma unroll
    for (int t = 0; t < 4; ++t) {
        int c = lane + 32 * t;
        float v = eluf(agg2[(long long)n * HID + c] + b2[c]);
        atomicAdd(&gsum[(long long)g * HID + c], v);
    }
    if (lane == 0) atomicAdd(&gcnt[g], 1.0f);
}

__global__ void k_div(const float* __restrict__ gsum, const float* __restrict__ gcnt,
                      float* __restrict__ out, int total) {
    int i = blockIdx.x * blockDim.x + threadIdx.x;
    if (i >= total) return;
    out[i] = gsum[i] / fmaxf(gcnt[i / HID], 1.0f);
}

// ---------------------------------------------------------------- launch
static inline int nblk(long long n, int b) { return (int)((n + b - 1) / b); }

extern "C" void kernel_launch(void* const* d_in, const int* in_sizes, int n_in,
                              void* d_out, int out_size, void* d_ws, size_t ws_size,
                              hipStream_t stream) {
    const float* x     = (const float*)d_in[0];
    const int*   ei    = (const int*)d_in[1];
    const int*   batch = (const int*)d_in[2];
    const float* W1    = (const float*)d_in[3];
    const float* asc1  = (const float*)d_in[4];
    const float* adt1  = (const float*)d_in[5];
    const float* b1    = (const float*)d_in[6];
    const float* W2    = (const float*)d_in[7];
    const float* asc2  = (const float*)d_in[8];
    const float* adt2  = (const float*)d_in[9];
    const float* b2    = (const float*)d_in[10];
    float* out = (float*)d_out;

    const int N  = in_sizes[0] / IN_F;
    const int E  = in_sizes[1] / 2;
    const int G  = out_size / HID;
    const int Et = E + N;

    // workspace carve (~320 MB total)
    float* p = (float*)d_ws;
    float* h1act = p; p += (size_t)N * L1OUT;
    float* h2lin = p; p += (size_t)N * HID;
    float* agg2  = p; p += (size_t)N * HID;
    float* xagg  = p; p += (size_t)N * HEADS * IN_F;
    float* as1n  = p; p += (size_t)N * HEADS;
    float* ad1n  = p; p += (size_t)N * HEADS;
    float* m1    = p; p += (size_t)N * HEADS;
    float* z1    = p; p += (size_t)N * HEADS;
    float* as2n  = p; p += N;
    float* ad2n  = p; p += N;
    float* m2    = p; p += N;
    float* z2    = p; p += N;
    float* gsum  = p; p += (size_t)G * HID;
    float* gcnt  = p; p += G;
    float* ws1   = p; p += HEADS * IN_F;
    float* wd1   = p; p += HEADS * IN_F;
    float* w2s   = p; p += L1OUT;
    float* w2d   = p; p += L1OUT;

    k_prep<<<1, 512, 0, stream>>>(W1, asc1, adt1, W2, asc2, adt2, ws1, wd1, w2s, w2d);

    k_fill<<<2048, 256, 0, stream>>>(xagg, (long long)N * HEADS * IN_F, 0.f);
    k_fill<<<2048, 256, 0, stream>>>(agg2, (long long)N * HID, 0.f);
    k_fill<<<1024, 256, 0, stream>>>(z1,   (long long)N * HEADS, 0.f);
    k_fill<<<1024, 256, 0, stream>>>(m1,   (long long)N * HEADS, -1e30f);
    k_fill<<<512,  256, 0, stream>>>(z2,   (long long)N, 0.f);
    k_fill<<<512,  256, 0, stream>>>(m2,   (long long)N, -1e30f);
    k_fill<<<512,  256, 0, stream>>>(gsum, (long long)G * HID, 0.f);
    k_fill<<<64,   256, 0, stream>>>(gcnt, (long long)G, 0.f);

    // ---- layer 1
    k_alpha1<<<nblk(N, 256), 256, 0, stream>>>(x, ws1, wd1, as1n, ad1n, N);
    k_edge_max<<<nblk(Et, 256), 256, 0, stream>>>(ei, E, N, as1n, ad1n, m1, HEADS);
    k_edge_sum<<<nblk(Et, 256), 256, 0, stream>>>(ei, E, N, as1n, ad1n, m1, z1, HEADS);
    k_scatter1<<<nblk(Et, 256), 256, 0, stream>>>(ei, E, N, x, as1n, ad1n, m1, z1, xagg);
    k_dense1<<<nblk(N, 8), 256, 0, stream>>>(xagg, W1, b1, w2s, w2d, h1act, as2n, ad2n, N);

    // ---- layer 2
    k_gemm2<<<nblk(N, 128), 256, 0, stream>>>(h1act, W2, h2lin, N);
    k_edge_max<<<nblk(Et, 256), 256, 0, stream>>>(ei, E, N, as2n, ad2n, m2, 1);
    k_edge_sum<<<nblk(Et, 256), 256, 0, stream>>>(ei, E, N, as2n, ad2n, m2, z2, 1);
    k_scatter2<<<nblk(Et, 8), 256, 0, stream>>>(ei, E, N, h2lin, as2n, ad2n, m2, z2, agg2);

    // ---- pool
    k_pool<<<nblk(N, 8), 256, 0, stream>>>(agg2, b2, batch, gsum, gcnt, N);
    k_div<<<nblk((long long)G * HID, 256), 256, 0, stream>>>(gsum, gcnt, out, G * HID);
}